// GCN_54348516164017
// MI455X (gfx1250) — compile-verified
//
#include <hip/hip_runtime.h>
#include <math.h>

typedef __attribute__((ext_vector_type(2))) float v2f;
typedef __attribute__((ext_vector_type(8))) float v8f;

#define F_IN  512
#define F_HID 64
#define F_OUT 40

// ------------------------------------------------------------------
// degree / normalization
// ------------------------------------------------------------------
__global__ __launch_bounds__(256) void k_init_dinv(float* __restrict__ dinv, int n) {
  int i = blockIdx.x * blockDim.x + threadIdx.x;
  if (i < n) dinv[i] = 1.0f;               // +1 for implicit self loop
}

__global__ __launch_bounds__(256) void k_count_deg(const long long* __restrict__ col,
                                                   float* __restrict__ dinv, int E) {
  int e = blockIdx.x * blockDim.x + threadIdx.x;
  if (e < E) atomicAdd(&dinv[(int)col[e]], 1.0f);
}

__global__ __launch_bounds__(256) void k_finalize_dinv(float* __restrict__ dinv, int n) {
  int i = blockIdx.x * blockDim.x + threadIdx.x;
  if (i < n) dinv[i] = rsqrtf(dinv[i]);
}

// ------------------------------------------------------------------
// GEMM1: h1[N x 64] = x[N x 512] @ W1[512 x 64], fp32 WMMA 16x16x4
// one wave -> 16 rows x full 64 cols (4 accumulators, A reused 4x)
// A layout (16x4 f32): lanes 0-15 row M, VGPR0=K0/K2, VGPR1=K1/K3
// B layout (4x16 f32): lanes stripe N, VGPR0=K0/K2, VGPR1=K1/K3
// C layout (16x16 f32): VGPR r -> row m0+r (lanes 0-15) / m0+r+8 (16-31)
// ------------------------------------------------------------------
__global__ __launch_bounds__(256) void k_gemm1(const float* __restrict__ X,
                                               const float* __restrict__ W,
                                               float* __restrict__ H, int M) {
  const int wave = blockIdx.x * (blockDim.x >> 5) + (threadIdx.x >> 5);
  const int lane = threadIdx.x & 31;
  const int m0 = wave << 4;
  if (m0 >= M) return;
  const int half = lane >> 4;   // 0: lanes 0-15, 1: lanes 16-31
  const int l16  = lane & 15;

  v8f a0 = {}, a1 = {}, a2 = {}, a3 = {};
  const float* xr = X + (size_t)(m0 + l16) * F_IN + 2 * half;

  for (int k = 0; k < F_IN; k += 4) {
    v2f A;
    A.x = xr[k];        // K = k + 2*half
    A.y = xr[k + 1];    // K = k + 2*half + 1
    const float* w0 = W + (size_t)(k + 2 * half) * F_HID + l16;
    const float* w1 = w0 + F_HID;
    v2f B0, B1, B2, B3;
    B0.x = w0[0];  B0.y = w1[0];
    B1.x = w0[16]; B1.y = w1[16];
    B2.x = w0[32]; B2.y = w1[32];
    B3.x = w0[48]; B3.y = w1[48];
    a0 = __builtin_amdgcn_wmma_f32_16x16x4_f32(false, A, false, B0, (short)0, a0, false, false);
    a1 = __builtin_amdgcn_wmma_f32_16x16x4_f32(false, A, false, B1, (short)0, a1, false, false);
    a2 = __builtin_amdgcn_wmma_f32_16x16x4_f32(false, A, false, B2, (short)0, a2, false, false);
    a3 = __builtin_amdgcn_wmma_f32_16x16x4_f32(false, A, false, B3, (short)0, a3, false, false);
  }

  for (int r = 0; r < 8; ++r) {
    int m = m0 + r + 8 * half;
    float* hp = H + (size_t)m * F_HID + l16;
    hp[0]  = a0[r];
    hp[16] = a1[r];
    hp[32] = a2[r];
    hp[48] = a3[r];
  }
}

// ------------------------------------------------------------------
// agg1 init: self-loop contribution + bias (ReLU deferred to GEMM2 A-load)
// ------------------------------------------------------------------
__global__ __launch_bounds__(256) void k_init_agg1(const float* __restrict__ h1,
                                                   const float* __restrict__ dinv,
                                                   const float* __restrict__ b1,
                                                   float* __restrict__ agg, int n) {
  int t = blockIdx.x * blockDim.x + threadIdx.x;
  if (t >= n * F_HID) return;
  int i = t >> 6;        // F_HID == 64
  int f = t & 63;
  float d = dinv[i];
  agg[t] = h1[t] * d * d + b1[f];
}

// ------------------------------------------------------------------
// edge scatter, layer 1: 64 lanes (features) per edge, coalesced
// ------------------------------------------------------------------
__global__ __launch_bounds__(256) void k_scatter1(const long long* __restrict__ row,
                                                  const long long* __restrict__ col,
                                                  const float* __restrict__ h,
                                                  const float* __restrict__ dinv,
                                                  float* __restrict__ agg, int E) {
  long long t = (long long)blockIdx.x * blockDim.x + threadIdx.x;
  int e = (int)(t >> 6);
  int f = (int)(t & 63);
  if (e >= E) return;
  int r = (int)row[e];
  int c = (int)col[e];
  float nrm = dinv[r] * dinv[c];
  atomicAdd(&agg[(size_t)c * F_HID + f], h[(size_t)r * F_HID + f] * nrm);
}

// ------------------------------------------------------------------
// GEMM2: h2[N x 40] = relu(agg1)[N x 64] @ W2[64 x 40], 40 cols padded to 48
// ------------------------------------------------------------------
__global__ __launch_bounds__(256) void k_gemm2(const float* __restrict__ Z,
                                               const float* __restrict__ W,
                                               float* __restrict__ H, int M) {
  const int wave = blockIdx.x * (blockDim.x >> 5) + (threadIdx.x >> 5);
  const int lane = threadIdx.x & 31;
  const int m0 = wave << 4;
  if (m0 >= M) return;
  const int half = lane >> 4;
  const int l16  = lane & 15;

  v8f a0 = {}, a1 = {}, a2 = {};
  const float* zr = Z + (size_t)(m0 + l16) * F_HID + 2 * half;
  const bool tail = (l16 < 8);   // cols 32..39 valid, 40..47 padded

  for (int k = 0; k < F_HID; k += 4) {
    v2f A;
    A.x = fmaxf(zr[k], 0.0f);        // fused ReLU
    A.y = fmaxf(zr[k + 1], 0.0f);
    const float* w0 = W + (size_t)(k + 2 * half) * F_OUT + l16;
    const float* w1 = w0 + F_OUT;
    v2f B0, B1, B2;
    B0.x = w0[0];  B0.y = w1[0];
    B1.x = w0[16]; B1.y = w1[16];
    B2.x = tail ? w0[32] : 0.0f;
    B2.y = tail ? w1[32] : 0.0f;
    a0 = __builtin_amdgcn_wmma_f32_16x16x4_f32(false, A, false, B0, (short)0, a0, false, false);
    a1 = __builtin_amdgcn_wmma_f32_16x16x4_f32(false, A, false, B1, (short)0, a1, false, false);
    a2 = __builtin_amdgcn_wmma_f32_16x16x4_f32(false, A, false, B2, (short)0, a2, false, false);
  }

  for (int r = 0; r < 8; ++r) {
    int m = m0 + r + 8 * half;
    float* hp = H + (size_t)m * F_OUT;
    hp[l16]      = a0[r];
    hp[16 + l16] = a1[r];
    if (tail) hp[32 + l16] = a2[r];
  }
}

// ------------------------------------------------------------------
// out init: self-loop + bias for layer 2, written straight into d_out
// ------------------------------------------------------------------
__global__ __launch_bounds__(256) void k_init_out(const float* __restrict__ h2,
                                                  const float* __restrict__ dinv,
                                                  const float* __restrict__ b2,
                                                  float* __restrict__ out, int n) {
  unsigned t = blockIdx.x * blockDim.x + threadIdx.x;
  if (t >= (unsigned)n * (unsigned)F_OUT) return;
  unsigned i = t / (unsigned)F_OUT;
  unsigned f = t - i * (unsigned)F_OUT;
  float d = dinv[i];
  out[t] = h2[t] * d * d + b2[f];
}

// ------------------------------------------------------------------
// edge scatter, layer 2: 40 features per edge, directly into d_out
// ------------------------------------------------------------------
__global__ __launch_bounds__(256) void k_scatter2(const long long* __restrict__ row,
                                                  const long long* __restrict__ col,
                                                  const float* __restrict__ h2,
                                                  const float* __restrict__ dinv,
                                                  float* __restrict__ out, int E) {
  unsigned t = blockIdx.x * blockDim.x + threadIdx.x;   // <= E*40 = 64M < 2^31
  if (t >= (unsigned)E * (unsigned)F_OUT) return;
  unsigned e = t / (unsigned)F_OUT;
  unsigned f = t - e * (unsigned)F_OUT;
  int r = (int)row[e];
  int c = (int)col[e];
  float nrm = dinv[r] * dinv[c];
  atomicAdd(&out[(size_t)c * F_OUT + f], h2[(size_t)r * F_OUT + f] * nrm);
}

// ------------------------------------------------------------------
// in-place log-softmax over 40 classes: one wave32 per node
// ------------------------------------------------------------------
__global__ __launch_bounds__(256) void k_logsoftmax(float* __restrict__ out, int n) {
  int wave = blockIdx.x * (blockDim.x >> 5) + (threadIdx.x >> 5);
  int lane = threadIdx.x & 31;
  if (wave >= n) return;
  float* rowp = out + (size_t)wave * F_OUT;
  float a = rowp[lane];
  float b = (lane < 8) ? rowp[32 + lane] : -__builtin_inff();
  float m = fmaxf(a, b);
  for (int off = 16; off; off >>= 1) m = fmaxf(m, __shfl_xor(m, off, 32));
  float s = expf(a - m) + ((lane < 8) ? expf(b - m) : 0.0f);
  for (int off = 16; off; off >>= 1) s += __shfl_xor(s, off, 32);
  float lse = logf(s) + m;
  rowp[lane] = a - lse;
  if (lane < 8) rowp[32 + lane] = b - lse;
}

// ------------------------------------------------------------------
extern "C" void kernel_launch(void* const* d_in, const int* in_sizes, int n_in,
                              void* d_out, int out_size, void* d_ws, size_t ws_size,
                              hipStream_t stream) {
  const float*     x   = (const float*)d_in[0];
  const long long* ei  = (const long long*)d_in[1];   // int64 [2, E]
  const float*     W1  = (const float*)d_in[2];
  const float*     b1  = (const float*)d_in[3];
  const float*     W2  = (const float*)d_in[4];
  const float*     b2  = (const float*)d_in[5];
  float*           out = (float*)d_out;

  const int N = in_sizes[0] / F_IN;   // 100000
  const int E = in_sizes[1] / 2;      // 1600000
  const long long* rowi = ei;
  const long long* coli = ei + E;

  // workspace carve-out (floats), 256B-aligned chunks
  float* wsf = (float*)d_ws;
  size_t o = 0;
  float* dinv = wsf + o;  o += ((size_t)N + 63) & ~(size_t)63;
  float* h1   = wsf + o;  o += (size_t)N * F_HID;
  float* agg1 = wsf + o;  o += (size_t)N * F_HID;
  float* h2   = wsf + o;  o += (size_t)N * F_OUT;
  (void)ws_size; (void)n_in; (void)out_size;

  const dim3 B(256);

  // normalization
  k_init_dinv    <<<(N + 255) / 256, B, 0, stream>>>(dinv, N);
  k_count_deg    <<<(E + 255) / 256, B, 0, stream>>>(coli, dinv, E);
  k_finalize_dinv<<<(N + 255) / 256, B, 0, stream>>>(dinv, N);

  // layer 1
  const int mt = (N + 15) / 16;                 // 6250 M-tiles
  const int gb = (mt + 7) / 8;                  // 8 waves / block
  k_gemm1    <<<gb, B, 0, stream>>>(x, W1, h1, N);
  k_init_agg1<<<(int)(((size_t)N * F_HID + 255) / 256), B, 0, stream>>>(h1, dinv, b1, agg1, N);
  k_scatter1 <<<(int)(((size_t)E * F_HID + 255) / 256), B, 0, stream>>>(rowi, coli, h1, dinv, agg1, E);

  // layer 2 (ReLU fused into GEMM2 A-load)
  k_gemm2    <<<gb, B, 0, stream>>>(agg1, W2, h2, N);
  k_init_out <<<(int)(((size_t)N * F_OUT + 255) / 256), B, 0, stream>>>(h2, dinv, b2, out, N);
  k_scatter2 <<<(int)(((size_t)E * F_OUT + 255) / 256), B, 0, stream>>>(rowi, coli, h2, dinv, out, E);

  // log-softmax, one wave per node
  k_logsoftmax<<<(N + 7) / 8, B, 0, stream>>>(out, N);
}